// IntraAttention_25237227831491
// MI455X (gfx1250) — compile-verified
//
#include <hip/hip_runtime.h>

typedef __attribute__((ext_vector_type(2))) float v2f;
typedef __attribute__((ext_vector_type(8))) float v8f;

#define BS    32
#define NSRC  4096
#define DIM   512
#define THIST 16

// Workspace layout (floats):
//   proj : [0,      16384)   = h_t @ W^T           [BS, DIM]
//   Cws  : [16384,  32768)   = sum_n w*h partials  [BS, DIM]
//   Sws  : [32768,  32800)   = sum_n w             [BS]
//
// Output layout (floats), concatenated in reference return order:
//   C_t               : [0,       16384)
//   alpha             : [16384,   147456)
//   next_attn_history : [147456,  2375680)   (first 16 slices = copy of input,
//                                             last slice = E)

__global__ void zero_ws_kernel(float* __restrict__ ws) {
    int i = blockIdx.x * blockDim.x + threadIdx.x;
    if (i < DIM * BS + BS) ws[16384 + i] = 0.0f;   // Cws + Sws
}

// -------- Kernel 1: proj = h_t @ W^T via V_WMMA_F32_16X16X4_F32 --------
// One wave per 16x16 output tile. A = h_t tile (16xK), B = W^T tile (Kx16),
// i.e. B[k][n] = W[n][k] so both operands load row-contiguous float2's.
__global__ __launch_bounds__(32) void proj_wmma_kernel(
    const float* __restrict__ h_t,   // [BS, DIM]
    const float* __restrict__ W,     // [DIM, DIM]
    float* __restrict__ proj)        // [BS, DIM]
{
    const int m0   = blockIdx.x * 16;      // 0 or 16   (rows of proj = b)
    const int n0   = blockIdx.y * 16;      // 0..496    (cols of proj)
    const int lane = threadIdx.x;
    const int half = lane >> 4;            // 0: K=k+0..1, 1: K=k+2..3
    const int l    = lane & 15;

    v8f acc = {};
    const float* arow = h_t + (size_t)(m0 + l) * DIM + 2 * half;
    const float* brow = W   + (size_t)(n0 + l) * DIM + 2 * half;

#pragma unroll 8
    for (int k = 0; k < DIM; k += 4) {
        v2f a = *(const v2f*)(arow + k);   // A[M=l][K=k+2*half, k+2*half+1]
        v2f b = *(const v2f*)(brow + k);   // B[K=k+2*half..][N=l] = W[n0+l][k..]
        acc = __builtin_amdgcn_wmma_f32_16x16x4_f32(
            /*neg_a=*/false, a, /*neg_b=*/false, b,
            /*c_mod=*/(short)0, acc, /*reuse_a=*/false, /*reuse_b=*/false);
    }

    // D layout: VGPR r holds row M = r (lanes 0-15) / M = 8+r (lanes 16-31)
#pragma unroll
    for (int r = 0; r < 8; ++r)
        proj[(size_t)(m0 + 8 * half + r) * DIM + n0 + l] = acc[r];
}

// -------- Kernel 2: fused E -> history-softmax -> weighted C accumulation ----
// One pass over h. Grid: (b, chunk) = (32, 16); 4 waves/WG; each wave owns one
// n at a time, lanes span the 512-wide d dimension (4x float4 per lane).
__global__ __launch_bounds__(128) void fused_attn_kernel(
    const float* __restrict__ h,          // [N, BS, DIM]
    const float* __restrict__ attn_hist,  // [T, BS, N]
    const float* __restrict__ proj,       // [BS, DIM]
    float* __restrict__ Cws,              // [BS, DIM] accum
    float* __restrict__ Sws,              // [BS] accum
    float* __restrict__ alpha_out,        // d_out alpha slab (gets w = E2)
    float* __restrict__ e_out)            // d_out history tail (gets E)
{
    const int b     = blockIdx.x;
    const int chunk = blockIdx.y;           // 256 n's per chunk
    const int wave  = threadIdx.x >> 5;
    const int lane  = threadIdx.x & 31;

    __shared__ float sC[DIM];
    __shared__ float sS;
    for (int i = threadIdx.x; i < DIM; i += blockDim.x) sC[i] = 0.0f;
    if (threadIdx.x == 0) sS = 0.0f;
    __syncthreads();

    // proj[b,:] resident in registers, same d-mapping as h loads
    float pj[16];
#pragma unroll
    for (int q = 0; q < 4; ++q) {
        const float4 p = *(const float4*)(proj + (size_t)b * DIM + q * 128 + lane * 4);
        pj[q * 4 + 0] = p.x; pj[q * 4 + 1] = p.y;
        pj[q * 4 + 2] = p.z; pj[q * 4 + 3] = p.w;
    }

    float Cacc[16];
#pragma unroll
    for (int i = 0; i < 16; ++i) Cacc[i] = 0.0f;
    float Ssum = 0.0f;

    const int t16 = lane & 15;  // lanes 0-15 and 16-31 duplicate the 16 t's
    const float* histp = attn_hist + (size_t)(t16 * BS + b) * NSRC;

    for (int i = 0; i < 64; ++i) {
        const int n = chunk * 256 + i * 4 + wave;
        const float* hp = h + ((size_t)n * BS + b) * DIM;

        if (i + 1 < 64)  // prefetch next row for this wave (64B/lane covers 2KB)
            __builtin_prefetch(h + ((size_t)(n + 4) * BS + b) * DIM + lane * 16, 0, 3);

        float hv[16];
        float e = 0.0f;
#pragma unroll
        for (int q = 0; q < 4; ++q) {
            const float4 x = *(const float4*)(hp + q * 128 + lane * 4);
            hv[q * 4 + 0] = x.x; hv[q * 4 + 1] = x.y;
            hv[q * 4 + 2] = x.z; hv[q * 4 + 3] = x.w;
            e += pj[q * 4 + 0] * x.x + pj[q * 4 + 1] * x.y +
                 pj[q * 4 + 2] * x.z + pj[q * 4 + 3] * x.w;
        }
        // wave32 all-reduce: E[b,n]
#pragma unroll
        for (int off = 16; off >= 1; off >>= 1) e += __shfl_xor(e, off, 32);

        // history max + denominator (each t loaded twice -> halve the sum)
        const float hval = histp[n];
        float m = hval;
#pragma unroll
        for (int off = 16; off >= 1; off >>= 1) m = fmaxf(m, __shfl_xor(m, off, 32));
        const float M = fmaxf(m, e);
        float ex = __expf(hval - M);
#pragma unroll
        for (int off = 16; off >= 1; off >>= 1) ex += __shfl_xor(ex, off, 32);
        const float w = __expf(e - M) / (0.5f * ex);   // E2[b,n]

        if (lane == 0) {
            e_out[(size_t)b * NSRC + n]     = e;   // next_attn_history[T]
            alpha_out[(size_t)b * NSRC + n] = w;   // normalized in finalize
        }
        Ssum += w;
#pragma unroll
        for (int j = 0; j < 16; ++j) Cacc[j] += w * hv[j];
    }

    // cross-wave reduction in LDS (ds_add_f32), then per-b global accumulation
#pragma unroll
    for (int q = 0; q < 4; ++q)
#pragma unroll
        for (int j = 0; j < 4; ++j)
            atomicAdd(&sC[q * 128 + lane * 4 + j], Cacc[q * 4 + j]);
    if (lane == 0) atomicAdd(&sS, Ssum);
    __syncthreads();

    for (int d = threadIdx.x; d < DIM; d += blockDim.x)
        atomicAdd(&Cws[(size_t)b * DIM + d], sC[d]);
    if (threadIdx.x == 0) atomicAdd(&Sws[b], sS);
}

// -------- Kernel 3: normalize C_t & alpha, copy history prefix ---------------
__global__ void finalize_kernel(
    const float* __restrict__ attn_hist,
    const float* __restrict__ Cws,
    const float* __restrict__ Sws,
    float* __restrict__ out)
{
    const long long tid = (long long)blockIdx.x * blockDim.x + threadIdx.x;
    const long long NC  = (long long)BS * DIM;            // 16384
    const long long NA  = (long long)BS * NSRC;           // 131072
    const long long NH4 = (long long)THIST * BS * NSRC/4; // 524288 float4's

    if (tid < NC) {
        const int b = (int)(tid >> 9);
        out[tid] = Cws[tid] / Sws[b];                     // C_t
    } else if (tid < NC + NA) {
        const long long i = tid - NC;
        const int b = (int)(i >> 12);
        out[NC + i] = out[NC + i] / Sws[b];               // alpha = E2 / sum
    } else if (tid < NC + NA + NH4) {
        const long long i = tid - NC - NA;
        float4* dst = (float4*)(out + NC + NA);
        const float4* src = (const float4*)attn_hist;
        dst[i] = src[i];                                  // history copy
    }
}

extern "C" void kernel_launch(void* const* d_in, const int* in_sizes, int n_in,
                              void* d_out, int out_size, void* d_ws, size_t ws_size,
                              hipStream_t stream) {
    const float* h_t       = (const float*)d_in[0];
    const float* h         = (const float*)d_in[1];
    const float* attn_hist = (const float*)d_in[2];
    const float* W         = (const float*)d_in[3];
    float* out = (float*)d_out;
    float* ws  = (float*)d_ws;

    float* proj = ws;
    float* Cws  = ws + 16384;
    float* Sws  = ws + 32768;

    float* alpha_out = out + 16384;                         // [BS, N]
    float* e_out     = out + 16384 + 131072 + THIST * BS * NSRC; // history tail

    zero_ws_kernel<<<65, 256, 0, stream>>>(ws);
    proj_wmma_kernel<<<dim3(2, 32), 32, 0, stream>>>(h_t, W, proj);
    fused_attn_kernel<<<dim3(32, 16), 128, 0, stream>>>(
        h, attn_hist, proj, Cws, Sws, alpha_out, e_out);
    finalize_kernel<<<2624, 256, 0, stream>>>(attn_hist, Cws, Sws, out);
}